// GHMC_21406117003629
// MI455X (gfx1250) — compile-verified
//
#include <hip/hip_runtime.h>
#include <hip/hip_bf16.h>
#include <stdint.h>

// ---------------- configuration ----------------
#define GHMC_BINS       30
#define GHMC_C          128
#define ROWS_PER_TILE   56          // 2 * 56*128*4 = 57344 B LDS for tiles (< 64KB static cap)
#define THREADS_A       256         // 8 wave32
#define WAVES_A         (THREADS_A / 32)
#define ROWS_PER_WAVE   (ROWS_PER_TILE / WAVES_A)   // 7
#define NBLOCKS_A       2048

// ---------------- TDM descriptor (CDNA5 ISA ch.8) ----------------
typedef unsigned int tdm_u32x4 __attribute__((ext_vector_type(4)));
typedef int          tdm_i32x8 __attribute__((ext_vector_type(8)));
typedef int          tdm_i32x4 __attribute__((ext_vector_type(4)));

// Issue a TDM load of `rows` x 128 fp32 tile starting at gptr into LDS byte
// offset ldsByteOff.  2D descriptor: dim0 = within-row (contig), dim1 = rows.
__device__ __forceinline__ void tdm_load_rows(const float* gptr,
                                              unsigned ldsByteOff,
                                              int rows, int nRowsTensor)
{
    unsigned long long ga = (unsigned long long)(uintptr_t)gptr;

    tdm_u32x4 g0;
    g0[0] = 1u;                                       // count=1 (valid), user mode
    g0[1] = ldsByteOff;                               // lds_addr (bytes)
    g0[2] = (unsigned)(ga & 0xFFFFFFFFull);           // global_addr[31:0]
    g0[3] = (unsigned)((ga >> 32) & 0x01FFFFFFull)    // global_addr[56:32]
          | (2u << 30);                               // type = 2 ("image")

    tdm_i32x8 g1;
    g1[0] = 0x20000;                                  // data_size=2 (4 bytes); no mask/flags
    g1[1] = (int)(((unsigned)GHMC_C & 0xFFFFu) << 16);            // tensor_dim0 lo16
    g1[2] = (int)((((unsigned)GHMC_C >> 16) & 0xFFFFu)            // tensor_dim0 hi16
          | (((unsigned)nRowsTensor & 0xFFFFu) << 16));           // tensor_dim1 lo16
    g1[3] = (int)((((unsigned)nRowsTensor >> 16) & 0xFFFFu)       // tensor_dim1 hi16
          | (((unsigned)GHMC_C & 0xFFFFu) << 16));                // tile_dim0 = 128
    g1[4] = rows & 0xFFFF;                            // tile_dim1 = rows, tile_dim2 = 0
    g1[5] = GHMC_C;                                   // tensor_dim0_stride lo32 = 128
    g1[6] = 0;                                        // stride hi / dim1_stride lo
    g1[7] = 0;                                        // dim1_stride hi

    tdm_i32x4 gz4; gz4[0] = 0; gz4[1] = 0; gz4[2] = 0; gz4[3] = 0;
    tdm_i32x8 gz8;
    gz8[0] = 0; gz8[1] = 0; gz8[2] = 0; gz8[3] = 0;
    gz8[4] = 0; gz8[5] = 0; gz8[6] = 0; gz8[7] = 0;

    // clang-23 / amdgpu-toolchain 6-arg form:
    // (g0 v4u, g1 v8i, g2 v4i, g3 v4i, extra v8i, cpol)
    __builtin_amdgcn_tensor_load_to_lds(g0, g1, gz4, gz4, gz8, 0);
}

// ---------------- pass 1: stream pred, accumulate per-bin {count, sum(-logp)} ----------------
__global__ __launch_bounds__(THREADS_A)
void ghmc_pass1(const float* __restrict__ pred,
                const float* __restrict__ target,
                const int*   __restrict__ label,
                float* __restrict__ partCnt,
                float* __restrict__ partSum,
                int nRows, int nTiles)
{
    __shared__ __align__(16) float tile[2][ROWS_PER_TILE * GHMC_C];
    __shared__ float    sS[GHMC_BINS];
    __shared__ unsigned sC[GHMC_BINS];

    const int tid  = threadIdx.x;
    const int lane = tid & 31;
    const int wv   = tid >> 5;

    if (tid < GHMC_BINS) { sS[tid] = 0.0f; sC[tid] = 0u; }

    const unsigned ldsBase0 = (unsigned)(uintptr_t)(&tile[0][0]);
    const unsigned ldsBase1 = (unsigned)(uintptr_t)(&tile[1][0]);

    const int t0 = blockIdx.x;
    // prologue: wave 0 kicks off the first TDM tile
    if (wv == 0 && t0 < nTiles) {
        int rows0 = nRows - t0 * ROWS_PER_TILE;
        if (rows0 > ROWS_PER_TILE) rows0 = ROWS_PER_TILE;
        tdm_load_rows(pred + (size_t)t0 * ROWS_PER_TILE * GHMC_C, ldsBase0, rows0, nRows);
    }
    __syncthreads();                         // bins initialized

    int buf = 0;
    for (int t = t0; t < nTiles; t += gridDim.x) {
        if (wv == 0) __builtin_amdgcn_s_wait_tensorcnt(0);
        __syncthreads();                     // tile[buf] visible to all waves

        const int tn = t + gridDim.x;        // prefetch next tile into other buffer
        if (wv == 0 && tn < nTiles) {
            int rowsN = nRows - tn * ROWS_PER_TILE;
            if (rowsN > ROWS_PER_TILE) rowsN = ROWS_PER_TILE;
            tdm_load_rows(pred + (size_t)tn * ROWS_PER_TILE * GHMC_C,
                          buf ? ldsBase0 : ldsBase1, rowsN, nRows);
        }

        int rowsHere = nRows - t * ROWS_PER_TILE;
        if (rowsHere > ROWS_PER_TILE) rowsHere = ROWS_PER_TILE;
        const size_t rowBase = (size_t)t * ROWS_PER_TILE;

        for (int i = 0; i < ROWS_PER_WAVE; ++i) {
            const int rLoc = wv * ROWS_PER_WAVE + i;
            if (rLoc >= rowsHere) break;
            const size_t gRow = rowBase + rLoc;

            // one 512B row per wave: 32 lanes x float4  (ds_load_b128)
            const float4 v =
                reinterpret_cast<const float4*>(&tile[buf][rLoc * GHMC_C])[lane];

            // wave32 max-reduce
            float m = fmaxf(fmaxf(v.x, v.y), fmaxf(v.z, v.w));
            #pragma unroll
            for (int off = 16; off >= 1; off >>= 1)
                m = fmaxf(m, __shfl_xor(m, off));

            // wave32 sum of exp
            float s = __expf(v.x - m) + __expf(v.y - m)
                    + __expf(v.z - m) + __expf(v.w - m);
            #pragma unroll
            for (int off = 16; off >= 1; off >>= 1)
                s += __shfl_xor(s, off);

            // labeled-class logit: owner lane = lab>>2, slot = lab&3
            int lab = 0;
            if (lane == 0) lab = label[gRow];
            lab = __shfl(lab, 0);
            const int slot = lab & 3;
            const float vsel = (slot == 0) ? v.x
                             : (slot == 1) ? v.y
                             : (slot == 2) ? v.z : v.w;
            const float pl = __shfl(vsel, lab >> 2);

            const float nlp = (m + __logf(s)) - pl;       // -log softmax at label

            if (lane == 0) {
                const float tg = target[gRow * GHMC_C + (size_t)lab];
                const float sg = 1.0f / (1.0f + __expf(-pl));
                const float g  = fabsf(sg - tg);
                int b = (int)(g * (float)GHMC_BINS);       // floor, g >= 0
                b = b < 0 ? 0 : (b > GHMC_BINS - 1 ? GHMC_BINS - 1 : b);
                atomicAdd(&sC[b], 1u);                     // ds_add_u32
                atomicAdd(&sS[b], nlp);                    // ds_add_f32
            }
        }
        __syncthreads();                     // all waves done reading tile[buf]
        buf ^= 1;
    }

    __syncthreads();
    if (tid < GHMC_BINS) {
        partCnt[(size_t)blockIdx.x * GHMC_BINS + tid] = (float)sC[tid];
        partSum[(size_t)blockIdx.x * GHMC_BINS + tid] = sS[tid];
    }
}

// ---------------- pass 2: deterministic reduce + finalize ----------------
// loss = (1/n_nonempty) * sum_b S_b / counts_b
__global__ __launch_bounds__(1024)
void ghmc_pass2(const float* __restrict__ partCnt,
                const float* __restrict__ partSum,
                int nPart, float* __restrict__ out)
{
    __shared__ float fc[32], fs[32];
    const int tid = threadIdx.x, lane = tid & 31, wv = tid >> 5;

    if (wv < GHMC_BINS) {                    // one wave per bin
        float c = 0.0f, s = 0.0f;
        for (int i = lane; i < nPart; i += 32) {
            c += partCnt[(size_t)i * GHMC_BINS + wv];
            s += partSum[(size_t)i * GHMC_BINS + wv];
        }
        #pragma unroll
        for (int off = 16; off >= 1; off >>= 1) {
            c += __shfl_xor(c, off);
            s += __shfl_xor(s, off);
        }
        if (lane == 0) { fc[wv] = c; fs[wv] = s; }
    }
    __syncthreads();

    if (wv == 0) {
        float c = (lane < GHMC_BINS) ? fc[lane] : 0.0f;
        float s = (lane < GHMC_BINS) ? fs[lane] : 0.0f;
        float nn   = (c > 0.0f) ? 1.0f : 0.0f;
        float term = (c > 0.0f) ? s / c : 0.0f;
        #pragma unroll
        for (int off = 16; off >= 1; off >>= 1) {
            nn   += __shfl_xor(nn, off);
            term += __shfl_xor(term, off);
        }
        if (lane == 0) out[0] = term / fmaxf(nn, 1.0f);
    }
}

// ---------------- launcher ----------------
extern "C" void kernel_launch(void* const* d_in, const int* in_sizes, int n_in,
                              void* d_out, int out_size, void* d_ws, size_t ws_size,
                              hipStream_t stream)
{
    (void)n_in; (void)out_size;
    const float* pred   = (const float*)d_in[0];
    const float* target = (const float*)d_in[1];
    const int*   label  = (const int*)d_in[2];
    float* out = (float*)d_out;

    const int nRows  = in_sizes[2];                          // N = 1048576
    const int nTiles = (nRows + ROWS_PER_TILE - 1) / ROWS_PER_TILE;

    int grid = NBLOCKS_A;
    if (grid > nTiles) grid = nTiles;
    // shrink grid if workspace is tight (needs grid*30*2 floats)
    while ((size_t)grid * GHMC_BINS * 2 * sizeof(float) > ws_size && grid > 1)
        grid >>= 1;

    float* partCnt = (float*)d_ws;
    float* partSum = partCnt + (size_t)grid * GHMC_BINS;

    ghmc_pass1<<<grid, THREADS_A, 0, stream>>>(pred, target, label,
                                               partCnt, partSum, nRows, nTiles);
    ghmc_pass2<<<1, 1024, 0, stream>>>(partCnt, partSum, grid, out);
}